// RENRG_89051851915728
// MI455X (gfx1250) — compile-verified
//
#include <hip/hip_runtime.h>
#include <cstddef>

// ============================================================================
// REN forward pass for MI455X (gfx1250). All heavy matrix algebra is routed
// through V_WMMA_F32_16X16X4_F32 (fp32 matrix cores). Matrix inverses are
// Newton-Schulz iterations (pure GEMMs -> WMMA). Workspace requirement:
// ~163 MB (see layout in kernel_launch).
// N_IN=512, M_OUT=1024, N_XI=1024, L=4096, DIM=6144.
// ============================================================================

#define EPS_REG 0.001f
#define NEWTON_ITERS 40

typedef float v2f __attribute__((ext_vector_type(2)));
typedef float v8f __attribute__((ext_vector_type(8)));

// ---------------------------------------------------------------------------
// Generic tiled fp32 WMMA GEMM:  C = alpha * op(A) @ op(B) + beta*C + diag*I
//   op(A): transA=0 -> A is MxK (lda), transA=1 -> A stored KxM (lda), use A^T
//   op(B): transB=0 -> B is KxN (ldb), transB=1 -> B stored NxK (ldb), use B^T
//   alpha = alpha_c * {1, g, 1/g} per alpha_mode (g read from gptr on device)
//   diag  = diag_c  * {1, g}      per diag_mode, added where global row==col
// Tiling: 128x128 per workgroup (256 threads = 8 waves), 32x64 per wave
// (2x4 WMMA accumulators), BK=16 (4 WMMA K-steps per LDS tile).
// LDS tiles are m/n-major with K contiguous (row stride 18 floats, 8B aligned)
// so every v2f fragment is ONE aligned ds_load_b64, bank-conflict-free.
// All M, N used here are divisible by 128 and K by 16.
// ---------------------------------------------------------------------------
#define BM 128
#define BN 128
#define BK 16
#define LDK (BK + 2)          // 18 floats: even (b64 aligned), 18i mod 64 distinct

__global__ __launch_bounds__(256)
void wmma_gemm_f32(const float* __restrict__ A, int lda, int transA,
                   const float* __restrict__ B, int ldb, int transB,
                   float* __restrict__ C, int ldc,
                   int M, int N, int K,
                   float alpha_c, int alpha_mode,
                   float beta,
                   float diag_c, int diag_mode,
                   const float* __restrict__ gptr)
{
  (void)M; (void)N;
  __shared__ float As[BM][LDK];
  __shared__ float Bs[BN][LDK];

  const int tile_m = blockIdx.y * BM;
  const int tile_n = blockIdx.x * BN;
  const int tid  = threadIdx.x;          // 0..255 (8 waves of 32)
  const int wave = tid >> 5;
  const int lane = tid & 31;
  const int half = lane >> 4;            // 0: lanes 0-15, 1: lanes 16-31
  const int l16  = lane & 15;
  const int wm = (wave & 3) * 32;        // 4 waves along M
  const int wn = (wave >> 2) * 64;       // 2 waves along N

  v8f acc[2][4];
  for (int i = 0; i < 2; ++i)
    for (int j = 0; j < 4; ++j)
      for (int e = 0; e < 8; ++e) acc[i][j][e] = 0.0f;

  float ra[8], rb[8];                    // register-staged tile (global->reg->LDS)

  for (int k0 = 0; k0 < K; k0 += BK) {
    // ---- global -> registers (overlaps with previous tile's WMMA compute) ----
#pragma unroll
    for (int t = 0; t < 8; ++t) {
      const int idx = tid + t * 256;     // 2048 = BM*BK elements
      if (transA) {                      // A stored KxM: coalesce along m
        const int m = idx & 127, k = idx >> 7;
        ra[t] = A[(size_t)(k0 + k) * lda + (tile_m + m)];
      } else {                           // A stored MxK: coalesce along k
        const int k = idx & 15, m = idx >> 4;
        ra[t] = A[(size_t)(tile_m + m) * lda + (k0 + k)];
      }
    }
#pragma unroll
    for (int t = 0; t < 8; ++t) {
      const int idx = tid + t * 256;
      if (transB) {                      // B stored NxK: coalesce along k
        const int k = idx & 15, n = idx >> 4;
        rb[t] = B[(size_t)(tile_n + n) * ldb + (k0 + k)];
      } else {                           // B stored KxN: coalesce along n
        const int n = idx & 127, k = idx >> 7;
        rb[t] = B[(size_t)(k0 + k) * ldb + (tile_n + n)];
      }
    }
    // prefetch one K-tile ahead (gfx1250 global_prefetch_b8)
    if (k0 + BK < K) {
      __builtin_prefetch(transA ? &A[(size_t)(k0 + BK) * lda + tile_m + (tid & 127)]
                                : &A[(size_t)(tile_m + (tid >> 4)) * lda + k0 + BK],
                         0, 0);
      __builtin_prefetch(transB ? &B[(size_t)(tile_n + (tid >> 4)) * ldb + k0 + BK]
                                : &B[(size_t)(k0 + BK) * ldb + tile_n + (tid & 127)],
                         0, 0);
    }
    __syncthreads();                     // previous compute done -> safe to overwrite LDS

    // ---- registers -> LDS (m/n-major, K contiguous) ----
#pragma unroll
    for (int t = 0; t < 8; ++t) {
      const int idx = tid + t * 256;
      if (transA) { As[idx & 127][idx >> 7] = ra[t]; }
      else        { As[idx >> 4][idx & 15]  = ra[t]; }
    }
#pragma unroll
    for (int t = 0; t < 8; ++t) {
      const int idx = tid + t * 256;
      if (transB) { Bs[idx >> 4][idx & 15]  = rb[t]; }
      else        { Bs[idx & 127][idx >> 7] = rb[t]; }
    }
    __syncthreads();

    // ---- WMMA compute: 8 per K-step, 32 per LDS tile ----
    // ISA 7.12.2, A 16x4 fp32: lanes0-15 hold {K0,K1}, lanes16-31 {K2,K3};
    // B mirrored (N across lanes). K-pair is contiguous -> single ds_load_b64.
#pragma unroll
    for (int kk = 0; kk < BK; kk += 4) {
      v2f af[2], bf[4];
#pragma unroll
      for (int i = 0; i < 2; ++i)
        af[i] = *(const v2f*)&As[wm + 16 * i + l16][kk + 2 * half];
#pragma unroll
      for (int j = 0; j < 4; ++j)
        bf[j] = *(const v2f*)&Bs[wn + 16 * j + l16][kk + 2 * half];
#pragma unroll
      for (int i = 0; i < 2; ++i)
#pragma unroll
        for (int j = 0; j < 4; ++j)
          acc[i][j] = __builtin_amdgcn_wmma_f32_16x16x4_f32(
              false, af[i], false, bf[j], (short)0, acc[i][j], false, false);
    }
  }

  const float g = gptr ? gptr[0] : 1.0f;
  const float alpha = alpha_c * (alpha_mode == 1 ? g : (alpha_mode == 2 ? 1.0f / g : 1.0f));
  const float diag  = diag_c  * (diag_mode  == 1 ? g : 1.0f);

  // C/D layout (ISA): VGPR v -> (M=v, N=lane) lanes0-15, (M=v+8, N=lane-16)
  for (int i = 0; i < 2; ++i)
    for (int j = 0; j < 4; ++j) {
      const int rbase = tile_m + wm + 16 * i + half * 8;
      const int c     = tile_n + wn + 16 * j + l16;
      for (int v = 0; v < 8; ++v) {
        const int r = rbase + v;
        const size_t off = (size_t)r * ldc + c;
        float out = alpha * acc[i][j][v];
        if (beta != 0.0f) out += beta * C[off];
        if (diag != 0.0f && r == c) out += diag;
        C[off] = out;
      }
    }
}

// ---------------------------------------------------------------------------
// GEMV: y = alpha * A(MxN, lda) @ x + beta*y.  One wave per row.
// ---------------------------------------------------------------------------
__global__ __launch_bounds__(256)
void gemv_kernel(const float* __restrict__ A, int lda, int M, int N,
                 const float* __restrict__ x, float* __restrict__ y,
                 float alpha, float beta)
{
  const int wave = threadIdx.x >> 5, lane = threadIdx.x & 31;
  const int row = blockIdx.x * 8 + wave;
  if (row >= M) return;
  const float* a = A + (size_t)row * lda;
  float p = 0.0f;
  for (int j = lane; j < N; j += 32) p += a[j] * x[j];
  for (int off = 16; off > 0; off >>= 1) p += __shfl_down(p, off, 32);
  if (lane == 0) y[row] = alpha * p + (beta != 0.0f ? beta * y[row] : 0.0f);
}

// ---------------------------------------------------------------------------
// Elementwise helpers
// ---------------------------------------------------------------------------
__global__ void build_A_kernel(const float* __restrict__ Mb, const float* __restrict__ Y3,
                               float* __restrict__ A, int n, float dadd)
{
  int idx = blockIdx.x * blockDim.x + threadIdx.x;
  if (idx >= n * n) return;
  int i = idx / n, j = idx - i * n;
  A[idx] = Mb[idx] + Y3[idx] - Y3[(size_t)j * n + i] + (i == j ? dadd : 0.0f);
}

__global__ void two_i_minus_kernel(const float* __restrict__ A, float* __restrict__ O, int n)
{
  int idx = blockIdx.x * blockDim.x + threadIdx.x;
  if (idx >= n * n) return;
  int i = idx / n, j = idx - i * n;
  O[idx] = (i == j ? 2.0f : 0.0f) - A[idx];
}

__global__ void build_E_kernel(const float* __restrict__ H11, const float* __restrict__ H33,
                               const float* __restrict__ Y, float* __restrict__ E, int n)
{
  int idx = blockIdx.x * blockDim.x + threadIdx.x;
  if (idx >= n * n) return;
  int i = idx / n, j = idx - i * n;
  E[idx] = 0.5f * (H11[idx] + H33[idx] + Y[idx] - Y[(size_t)j * n + i]);
}

__global__ void scale_copy_kernel(const float* __restrict__ s, float* __restrict__ d,
                                  float a, int cnt)
{
  int i = blockIdx.x * blockDim.x + threadIdx.x;
  if (i < cnt) d[i] = a * s[i];
}

// ---------------------------------------------------------------------------
// Newton-Schulz inversion helpers
// ---------------------------------------------------------------------------
__global__ void zero2_kernel(unsigned* p) { if (threadIdx.x < 2) p[threadIdx.x] = 0u; }

__global__ void row_abssum_max(const float* __restrict__ A, int n, unsigned* out)
{
  int r = blockIdx.x * blockDim.x + threadIdx.x;
  if (r >= n) return;
  const float* a = A + (size_t)r * n;
  float s = 0.0f;
  for (int j = 0; j < n; ++j) s += fabsf(a[j]);
  atomicMax(out, __float_as_uint(s));   // positive floats: uint order == float order
}

__global__ void col_abssum_max(const float* __restrict__ A, int n, unsigned* out)
{
  int c = blockIdx.x * blockDim.x + threadIdx.x;
  if (c >= n) return;
  float s = 0.0f;
  for (int i = 0; i < n; ++i) s += fabsf(A[(size_t)i * n + c]);
  atomicMax(out, __float_as_uint(s));
}

__global__ void newton_init_kernel(const float* __restrict__ A, float* __restrict__ X0,
                                   int n, const unsigned* __restrict__ s)
{
  int idx = blockIdx.x * blockDim.x + threadIdx.x;
  if (idx >= n * n) return;
  int i = idx / n, j = idx - i * n;
  float sc = __uint_as_float(s[0]) * __uint_as_float(s[1]);
  X0[idx] = A[(size_t)j * n + i] / sc;   // X0 = A^T / (||A||_inf * ||A||_1)
}

// ---------------------------------------------------------------------------
// Sequential epsilon scan (L=4096): eps[i] = relu((c[i] - H22[i,:i]@eps)/H22[i,i])
// Single workgroup, eps LDS-resident.
// ---------------------------------------------------------------------------
#define LSZ 4096
__global__ __launch_bounds__(1024)
void scan_kernel(const float* __restrict__ H22, const float* __restrict__ c,
                 float* __restrict__ eps_out)
{
  __shared__ float s_eps[LSZ];
  __shared__ float s_red[1024];
  const int tid = threadIdx.x;
  for (int i = tid; i < LSZ; i += 1024) s_eps[i] = 0.0f;
  __syncthreads();
  for (int i = 0; i < LSZ; ++i) {
    const float* row = H22 + (size_t)i * LSZ;
    float p = 0.0f;
    for (int j = tid; j < i; j += 1024) p += row[j] * s_eps[j];
    s_red[tid] = p;
    __syncthreads();
    for (int s = 512; s > 0; s >>= 1) {
      if (tid < s) s_red[tid] += s_red[tid + s];
      __syncthreads();
    }
    if (tid == 0) {
      float v = (c[i] - s_red[0]) / row[i];   // D11 = -tril(H22,-1), Lam = diag
      s_eps[i] = v > 0.0f ? v : 0.0f;
    }
    __syncthreads();
  }
  for (int i = tid; i < LSZ; i += 1024) eps_out[i] = s_eps[i];
}

// ---------------------------------------------------------------------------
// Host-side helpers
// ---------------------------------------------------------------------------
static inline void gemm(hipStream_t st, const float* A, int lda, int tA,
                        const float* B, int ldb, int tB,
                        float* C, int ldc, int M, int N, int K,
                        float alpha_c, int amode, float beta,
                        float diag_c, int dmode, const float* gptr)
{
  dim3 grid(N / BN, M / BM), block(256);
  wmma_gemm_f32<<<grid, block, 0, st>>>(A, lda, tA, B, ldb, tB, C, ldc,
                                        M, N, K, alpha_c, amode, beta,
                                        diag_c, dmode, gptr);
}

static inline void gemv(hipStream_t st, const float* A, int lda, int M, int N,
                        const float* x, float* y, float alpha, float beta)
{
  gemv_kernel<<<dim3(M / 8), dim3(256), 0, st>>>(A, lda, M, N, x, y, alpha, beta);
}

// Newton-Schulz inverse: Xdst = inv(A), scratch T and Xalt each >= n*n floats.
static void newton_invert(hipStream_t st, const float* A, float* Xdst,
                          float* T, float* Xalt, int n, unsigned* scal)
{
  const int nn = n * n, EW = 256;
  zero2_kernel<<<1, 32, 0, st>>>(scal);
  row_abssum_max<<<(n + 255) / 256, 256, 0, st>>>(A, n, scal + 0);
  col_abssum_max<<<(n + 255) / 256, 256, 0, st>>>(A, n, scal + 1);
  newton_init_kernel<<<(nn + EW - 1) / EW, EW, 0, st>>>(A, Xdst, n, scal);
  float* X = Xdst;
  float* Y = Xalt;
  for (int it = 0; it < NEWTON_ITERS; ++it) {   // even count -> ends in Xdst
    gemm(st, A, n, 0, X, n, 0, T, n, n, n, n, 1.0f, 0, 0.0f, 0.0f, 0, nullptr);
    two_i_minus_kernel<<<(nn + EW - 1) / EW, EW, 0, st>>>(T, T, n);
    gemm(st, X, n, 0, T, n, 0, Y, n, n, n, n, 1.0f, 0, 0.0f, 0.0f, 0, nullptr);
    float* tmp = X; X = Y; Y = tmp;
  }
}

// ---------------------------------------------------------------------------
extern "C" void kernel_launch(void* const* d_in, const int* in_sizes, int n_in,
                              void* d_out, int out_size, void* d_ws, size_t ws_size,
                              hipStream_t stream)
{
  (void)in_sizes; (void)n_in; (void)out_size; (void)ws_size;
  const int NIN = 512, MOUT = 1024, NXI = 1024, L = 4096, DIM = 6144;

  const float* X   = (const float*)d_in[0];   // DIM x DIM
  const float* Y   = (const float*)d_in[1];   // NXI x NXI
  const float* B2  = (const float*)d_in[2];   // NXI x NIN
  const float* C2  = (const float*)d_in[3];   // MOUT x NXI
  const float* D21 = (const float*)d_in[4];   // MOUT x L
  const float* Z3  = (const float*)d_in[5];   // (MOUT-NIN) x NIN = 512x512
  const float* X3  = (const float*)d_in[6];   // 512x512
  const float* Y3  = (const float*)d_in[7];   // 512x512
  const float* D12 = (const float*)d_in[8];   // L x NIN
  const float* wv  = (const float*)d_in[9];   // NIN
  const float* xi  = (const float*)d_in[10];  // NXI
  const float* gp  = (const float*)d_in[11];  // gammap (scalar)

  float* out_u  = (float*)d_out;              // u: MOUT
  float* out_xi = out_u + MOUT;               // xi_next: NXI

  // ---- workspace layout (floats). Total ~40.9M floats ~= 163 MB ----
  float* wsf = (float*)d_ws;
  size_t off = 0;
  auto alloc = [&](size_t n) { float* p = wsf + off; off += n; return p; };
  float* Mbuf  = alloc(512 * 512);
  float* Abuf  = alloc(512 * 512);
  float* IMbuf = alloc(512 * 512);
  float* Minv  = alloc(512 * 512);
  float* Rcap  = alloc(512 * 512);
  float* Rinv  = alloc(512 * 512);
  float* Einv  = alloc(1024 * 1024);
  float* P0    = alloc(1024 * 1024);          // Newton scratch T
  float* P1    = alloc(1024 * 1024);          // Newton scratch Xalt
  float* D22   = alloc((size_t)MOUT * NIN);
  float* vecR  = alloc((size_t)DIM * NIN);
  float* Wbuf  = alloc((size_t)DIM * NIN);
  float* Hb11  = alloc((size_t)NXI * NXI);
  float* Hb21  = alloc((size_t)L * NXI);
  float* Hb22  = alloc((size_t)L * L);
  float* Hb31  = alloc((size_t)NXI * NXI);
  float* Hb32  = alloc((size_t)NXI * L);
  float* Hb33  = alloc((size_t)NXI * NXI);
  float* Ebuf  = alloc((size_t)NXI * NXI);
  float* cvec  = alloc(L);
  float* epsv  = alloc(L);
  float* Exi   = alloc(NXI);
  unsigned* scal = (unsigned*)alloc(4);
  const int EW = 256;

  // ---- M = X3^T X3 + Z3^T Z3 (symmetric parts) ----
  gemm(stream, X3, 512, 1, X3, 512, 0, Mbuf, 512, 512, 512, 512, 1.0f, 0, 0.0f, 0.0f, 0, nullptr);
  gemm(stream, Z3, 512, 1, Z3, 512, 0, Mbuf, 512, 512, 512, 512, 1.0f, 0, 1.0f, 0.0f, 0, nullptr);
  // A = I + M  (adds Y3 - Y3^T + (1+eps) I)
  build_A_kernel<<<(512 * 512 + EW - 1) / EW, EW, 0, stream>>>(Mbuf, Y3, Abuf, 512, 1.0f + EPS_REG);
  // Minv = inv(I + M)
  newton_invert(stream, Abuf, Minv, P0, P1, 512, scal);
  // IM = I - M = 2I - A
  two_i_minus_kernel<<<(512 * 512 + EW - 1) / EW, EW, 0, stream>>>(Abuf, IMbuf, 512);
  // D22 = g * [ (I-M)Minv ; -2 Z3 Minv ]
  gemm(stream, IMbuf, 512, 0, Minv, 512, 0, D22, 512, 512, 512, 512, 1.0f, 1, 0.0f, 0.0f, 0, gp);
  gemm(stream, Z3, 512, 0, Minv, 512, 0, D22 + (size_t)512 * 512, 512, 512, 512, 512, -2.0f, 1, 0.0f, 0.0f, 0, gp);
  // R_cap = g I - (1/g) D22^T D22 ; Rinv = inv(R_cap)
  gemm(stream, D22, 512, 1, D22, 512, 0, Rcap, 512, 512, 512, 1024, -1.0f, 2, 0.0f, 1.0f, 1, gp);
  newton_invert(stream, Rcap, Rinv, P0, P1, 512, scal);

  // ---- vec_R = [ -(1/g) C2^T D22 ; -(1/g) D21^T D22 - D12 ; B2 ] ----
  gemm(stream, C2, 1024, 1, D22, 512, 0, vecR, 512, 1024, 512, 1024, -1.0f, 2, 0.0f, 0.0f, 0, gp);
  scale_copy_kernel<<<(L * NIN + EW - 1) / EW, EW, 0, stream>>>(D12, vecR + (size_t)1024 * 512, -1.0f, L * NIN);
  gemm(stream, D21, 4096, 1, D22, 512, 0, vecR + (size_t)1024 * 512, 512, 4096, 512, 1024, -1.0f, 2, 1.0f, 0.0f, 0, gp);
  scale_copy_kernel<<<(NXI * NIN + EW - 1) / EW, EW, 0, stream>>>(B2, vecR + (size_t)5120 * 512, 1.0f, NXI * NIN);
  // W = vec_R @ Rinv
  gemm(stream, vecR, 512, 0, Rinv, 512, 0, Wbuf, 512, DIM, 512, 512, 1.0f, 0, 0.0f, 0.0f, 0, nullptr);

  // ---- Needed blocks of H = X^T X + eps I + W vec_R^T + (1/g) vecQ vecQ^T ----
  const float* Xc0 = X;            // cols 0:1024
  const float* Xc1 = X + 1024;     // cols 1024:5120
  const float* Xc2 = X + 5120;     // cols 5120:6144
  float* W0 = Wbuf;                         float* R0 = vecR;
  float* W1 = Wbuf + (size_t)1024 * 512;    float* R1 = vecR + (size_t)1024 * 512;
  float* W2 = Wbuf + (size_t)5120 * 512;    float* R2 = vecR + (size_t)5120 * 512;

  // H11 (1024x1024)
  gemm(stream, Xc0, DIM, 1, Xc0, DIM, 0, Hb11, 1024, 1024, 1024, DIM, 1.0f, 0, 0.0f, EPS_REG, 0, nullptr);
  gemm(stream, W0, 512, 0, R0, 512, 1, Hb11, 1024, 1024, 1024, 512, 1.0f, 0, 1.0f, 0.0f, 0, nullptr);
  gemm(stream, C2, 1024, 1, C2, 1024, 0, Hb11, 1024, 1024, 1024, 1024, 1.0f, 2, 1.0f, 0.0f, 0, gp);
  // H21 (4096x1024)
  gemm(stream, Xc1, DIM, 1, Xc0, DIM, 0, Hb21, 1024, 4096, 1024, DIM, 1.0f, 0, 0.0f, 0.0f, 0, nullptr);
  gemm(stream, W1, 512, 0, R0, 512, 1, Hb21, 1024, 4096, 1024, 512, 1.0f, 0, 1.0f, 0.0f, 0, nullptr);
  gemm(stream, D21, 4096, 1, C2, 1024, 0, Hb21, 1024, 4096, 1024, 1024, 1.0f, 2, 1.0f, 0.0f, 0, gp);
  // H22 (4096x4096)
  gemm(stream, Xc1, DIM, 1, Xc1, DIM, 0, Hb22, 4096, 4096, 4096, DIM, 1.0f, 0, 0.0f, EPS_REG, 0, nullptr);
  gemm(stream, W1, 512, 0, R1, 512, 1, Hb22, 4096, 4096, 4096, 512, 1.0f, 0, 1.0f, 0.0f, 0, nullptr);
  gemm(stream, D21, 4096, 1, D21, 4096, 0, Hb22, 4096, 4096, 4096, 1024, 1.0f, 2, 1.0f, 0.0f, 0, gp);
  // H31 (1024x1024)
  gemm(stream, Xc2, DIM, 1, Xc0, DIM, 0, Hb31, 1024, 1024, 1024, DIM, 1.0f, 0, 0.0f, 0.0f, 0, nullptr);
  gemm(stream, W2, 512, 0, R0, 512, 1, Hb31, 1024, 1024, 1024, 512, 1.0f, 0, 1.0f, 0.0f, 0, nullptr);
  // H32 (1024x4096)
  gemm(stream, Xc2, DIM, 1, Xc1, DIM, 0, Hb32, 4096, 1024, 4096, DIM, 1.0f, 0, 0.0f, 0.0f, 0, nullptr);
  gemm(stream, W2, 512, 0, R1, 512, 1, Hb32, 4096, 1024, 4096, 512, 1.0f, 0, 1.0f, 0.0f, 0, nullptr);
  // H33 (1024x1024)
  gemm(stream, Xc2, DIM, 1, Xc2, DIM, 0, Hb33, 1024, 1024, 1024, DIM, 1.0f, 0, 0.0f, EPS_REG, 0, nullptr);
  gemm(stream, W2, 512, 0, R2, 512, 1, Hb33, 1024, 1024, 1024, 512, 1.0f, 0, 1.0f, 0.0f, 0, nullptr);

  // ---- E = 0.5 (H11 + H33 + Y - Y^T) ----
  build_E_kernel<<<(NXI * NXI + EW - 1) / EW, EW, 0, stream>>>(Hb11, Hb33, Y, Ebuf, NXI);

  // ---- c = -H21 @ xi + D12 @ w ; sequential ReLU scan for epsilon ----
  gemv(stream, Hb21, 1024, L, NXI, xi, cvec, -1.0f, 0.0f);
  gemv(stream, D12, 512, L, NIN, wv, cvec, 1.0f, 1.0f);
  scan_kernel<<<1, 1024, 0, stream>>>(Hb22, cvec, epsv);

  // ---- E_xi = H31 xi + H32 eps + B2 w ; xi_next = inv(E) E_xi ----
  gemv(stream, Hb31, 1024, NXI, NXI, xi, Exi, 1.0f, 0.0f);
  gemv(stream, Hb32, 4096, NXI, L, epsv, Exi, 1.0f, 1.0f);
  gemv(stream, B2, 512, NXI, NIN, wv, Exi, 1.0f, 1.0f);
  newton_invert(stream, Ebuf, Einv, P0, P1, 1024, scal);
  gemv(stream, Einv, 1024, NXI, NXI, Exi, out_xi, 1.0f, 0.0f);

  // ---- u = C2 xi + D21 eps + D22 w ----
  gemv(stream, C2, 1024, MOUT, NXI, xi, out_u, 1.0f, 0.0f);
  gemv(stream, D21, 4096, MOUT, L, epsv, out_u, 1.0f, 1.0f);
  gemv(stream, D22, 512, MOUT, NIN, wv, out_u, 1.0f, 1.0f);
}